// TripletLoss_65712999629573
// MI455X (gfx1250) — compile-verified
//
#include <hip/hip_runtime.h>
#include <hip/hip_bf16.h>
#include <float.h>

typedef float v2f __attribute__((ext_vector_type(2)));
typedef float v8f __attribute__((ext_vector_type(8)));

#define NROWS 4096
#define DIM   128
#define TILE  16
#define LDSSTRIDE (DIM + 4)              // 132 dwords -> conflict-free b64 reads
#define WAVES_PER_BLOCK 8
#define ROWS_PER_WAVE   32               // two 16-row M tiles per wave
#define ROWS_PER_BLOCK  (WAVES_PER_BLOCK * ROWS_PER_WAVE)   // 256

#if __has_builtin(__builtin_amdgcn_global_load_async_to_lds_b128) && \
    __has_builtin(__builtin_amdgcn_s_wait_asynccnt)
#define USE_ASYNC_LDS 1
#else
#define USE_ASYNC_LDS 0
#endif

#if USE_ASYNC_LDS
typedef int v4i __attribute__((vector_size(4 * sizeof(int))));
typedef __attribute__((address_space(1))) v4i g_v4i;   // global
typedef __attribute__((address_space(3))) v4i l_v4i;   // LDS
#endif

// ---------------------------------------------------------------------------
// Kernel A: per-row squared norms  sq[i] = sum_k X[i][k]^2
// ---------------------------------------------------------------------------
__global__ __launch_bounds__(256) void sumsq_kernel(const float* __restrict__ X,
                                                    float* __restrict__ sq) {
    int i = blockIdx.x * blockDim.x + threadIdx.x;
    if (i >= NROWS) return;
    const float4* xp = (const float4*)(X + (size_t)i * DIM);
    float s = 0.f;
#pragma unroll
    for (int k = 0; k < DIM / 4; ++k) {
        float4 v = xp[k];
        s += v.x * v.x + v.y * v.y + v.z * v.z + v.w * v.w;
    }
    sq[i] = s;
}

// ---------------------------------------------------------------------------
// Kernel B: WMMA Gram tiles + fused hardest-pos/neg mining.
// Each wave owns 32 rows (two M tiles sharing every B fragment).
// Column tiles are double-buffered in LDS via async global->LDS copies.
// ---------------------------------------------------------------------------
__global__ __launch_bounds__(256) void hardmine_kernel(
    const float* __restrict__ X, const int* __restrict__ tgt,
    const float* __restrict__ sq,
    float* __restrict__ dap, float* __restrict__ dan, int* __restrict__ aidx) {
    __shared__ float bt[2][TILE][LDSSTRIDE];

    const int tid  = threadIdx.x;
    const int lane = tid & 31;
    const int wave = tid >> 5;
    const int half = lane >> 4;   // K-half select (0/1)
    const int l16  = lane & 15;   // M (A) / N (B,C) index
    const int r0   = blockIdx.x * ROWS_PER_BLOCK + wave * ROWS_PER_WAVE;

    // Preload A fragments for the wave's two 16-row tiles.
    // A 16x4 f32 layout: lane(l16)=M; v0 holds K=k0+2*half, v1 holds K=k0+2*half+1.
    v2f afragA[DIM / 4], afragB[DIM / 4];
    {
        const float* xa = X + (size_t)(r0 + l16) * DIM + half * 2;
        const float* xb = X + (size_t)(r0 + TILE + l16) * DIM + half * 2;
#pragma unroll
        for (int s = 0; s < DIM / 4; ++s) {
            afragA[s].x = xa[4 * s + 0];
            afragA[s].y = xa[4 * s + 1];
            afragB[s].x = xb[4 * s + 0];
            afragB[s].y = xb[4 * s + 1];
        }
    }

    // Per-lane metadata for the 16 C rows this lane holds.
    // q in [0,8): tile0 row r0 + q + 8*half ; q in [8,16): tile1 row r0+16 + (q-8) + 8*half
    float sqi[16];
    int   ti[16];
#pragma unroll
    for (int q = 0; q < 16; ++q) {
        int row = r0 + (q & 8) * 2 + (q & 7) + 8 * half;
        sqi[q] = sq[row];
        ti[q]  = tgt[row];
    }
    float maxv[16], minv[16];
    int   amax[16];
#pragma unroll
    for (int q = 0; q < 16; ++q) { maxv[q] = -FLT_MAX; minv[q] = FLT_MAX; amax[q] = 0; }

    // staging coords: thread stages 8 floats of column row (tid>>4)
    const int tr = tid >> 4;
    const int tc = (tid & 15) * 8;

    // --- prologue: fill buffer 0 with column tile 0 ---
    {
        const float* src = X + (size_t)tr * DIM + tc;
        float* dst = &bt[0][tr][tc];
#if USE_ASYNC_LDS
        __builtin_amdgcn_global_load_async_to_lds_b128((g_v4i*)src,       (l_v4i*)dst,       0, 0);
        __builtin_amdgcn_global_load_async_to_lds_b128((g_v4i*)(src + 4), (l_v4i*)(dst + 4), 0, 0);
        __builtin_amdgcn_s_wait_asynccnt(0);
#else
        const float4* s4 = (const float4*)src;
        float4 a = s4[0], b = s4[1];
        *(float4*)(dst)     = a;
        *(float4*)(dst + 4) = b;
#endif
    }
    __syncthreads();

    const int NT = NROWS / TILE;   // 256 column tiles
    for (int t = 0; t < NT; ++t) {
        const int cur = t & 1;
        const int j0  = t * TILE;

        // prefetch next column tile into the other buffer (overlaps WMMA below)
        if (t + 1 < NT) {
            const float* src = X + (size_t)(j0 + TILE + tr) * DIM + tc;
            float* dst = &bt[cur ^ 1][tr][tc];
#if USE_ASYNC_LDS
            __builtin_amdgcn_global_load_async_to_lds_b128((g_v4i*)src,       (l_v4i*)dst,       0, 0);
            __builtin_amdgcn_global_load_async_to_lds_b128((g_v4i*)(src + 4), (l_v4i*)(dst + 4), 0, 0);
#else
            const float4* s4 = (const float4*)src;
            float4 a = s4[0], b = s4[1];
            *(float4*)(dst)     = a;
            *(float4*)(dst + 4) = b;
#endif
        }

        // --- 2 x 32 WMMAs: both M tiles share every B fragment ---
        v8f c0 = {0.f, 0.f, 0.f, 0.f, 0.f, 0.f, 0.f, 0.f};
        v8f c1 = {0.f, 0.f, 0.f, 0.f, 0.f, 0.f, 0.f, 0.f};
#pragma unroll
        for (int s = 0; s < DIM / 4; ++s) {
            v2f b;
            const float* bp = &bt[cur][l16][4 * s + half * 2];
            b.x = bp[0];
            b.y = bp[1];
            c0 = __builtin_amdgcn_wmma_f32_16x16x4_f32(false, afragA[s], false, b,
                                                       (short)0, c0, false, false);
            c1 = __builtin_amdgcn_wmma_f32_16x16x4_f32(false, afragB[s], false, b,
                                                       (short)0, c1, false, false);
        }

        // --- fold tile into running hardest-pos / hardest-neg state ---
        float sqj  = sq[j0 + l16];
        int   tj   = tgt[j0 + l16];
        int   jcol = j0 + l16;
#pragma unroll
        for (int q = 0; q < 16; ++q) {
            float dot  = (q < 8) ? c0[q & 7] : c1[q & 7];
            float dsq  = sqi[q] + sqj - 2.0f * dot;
            float dist = sqrtf(fmaxf(dsq, 1e-12f));
            if (ti[q] == tj) {
                if (dist >= maxv[q]) { maxv[q] = dist; amax[q] = jcol; }  // >= keeps LAST tie
            } else {
                minv[q] = fminf(minv[q], dist);
            }
        }

#if USE_ASYNC_LDS
        __builtin_amdgcn_s_wait_asynccnt(0);
#endif
        __syncthreads();
    }

    // Reduce across the 16 lanes sharing each C row (xor masks 1..8 stay in half).
#pragma unroll
    for (int q = 0; q < 16; ++q) {
        float mv = maxv[q];
        int   mi = amax[q];
        float nv = minv[q];
#pragma unroll
        for (int off = 1; off < 16; off <<= 1) {
            float omv = __shfl_xor(mv, off, 32);
            int   omi = __shfl_xor(mi, off, 32);
            float onv = __shfl_xor(nv, off, 32);
            if (omv > mv || (omv == mv && omi > mi)) { mv = omv; mi = omi; }  // last tie wins
            nv = fminf(nv, onv);
        }
        maxv[q] = mv; amax[q] = mi; minv[q] = nv;
    }
    if (l16 == 0) {
#pragma unroll
        for (int q = 0; q < 16; ++q) {
            int row = r0 + (q & 8) * 2 + (q & 7) + 8 * half;
            dap[row]  = maxv[q];
            dan[row]  = minv[q];
            aidx[row] = amax[q];
        }
    }
}

// ---------------------------------------------------------------------------
// Kernel C: per-row losses. an3 = dist[i, ind[i]], an2 = dist[lastap[i], ind[i]]
// ---------------------------------------------------------------------------
__global__ __launch_bounds__(256) void pairloss_kernel(
    const float* __restrict__ X, const float* __restrict__ sq,
    const int* __restrict__ ind,
    const float* __restrict__ dap, const float* __restrict__ dan,
    const int* __restrict__ aidx,
    float* __restrict__ loss, float* __restrict__ loss3) {
    int i = blockIdx.x * blockDim.x + threadIdx.x;
    if (i >= NROWS) return;
    int cidx = ind[i];
    int ridx = aidx[i];
    const float4* xi = (const float4*)(X + (size_t)i * DIM);
    const float4* xr = (const float4*)(X + (size_t)ridx * DIM);
    const float4* xc = (const float4*)(X + (size_t)cidx * DIM);
    float d1 = 0.f, d2 = 0.f;
#pragma unroll
    for (int k = 0; k < DIM / 4; ++k) {
        float4 a = xi[k], b = xr[k], cc = xc[k];
        d1 += a.x * cc.x + a.y * cc.y + a.z * cc.z + a.w * cc.w;
        d2 += b.x * cc.x + b.y * cc.y + b.z * cc.z + b.w * cc.w;
    }
    float an3 = sqrtf(fmaxf(sq[i]    + sq[cidx] - 2.f * d1, 1e-12f));
    float an2 = sqrtf(fmaxf(sq[ridx] + sq[cidx] - 2.f * d2, 1e-12f));
    loss[i]  = fmaxf(dap[i] - dan[i], 0.f);   // margin = 0
    loss3[i] = fabsf(an3 - an2);
}

// ---------------------------------------------------------------------------
// Kernel D: deterministic single-block reduction -> 3 scalars
// ---------------------------------------------------------------------------
__global__ __launch_bounds__(256) void finalize_kernel(
    const float* __restrict__ loss, const float* __restrict__ loss3,
    float* __restrict__ out) {
    __shared__ float s1[256], s2[256];
    int t = threadIdx.x;
    float a = 0.f, b = 0.f;
    for (int i = t; i < NROWS; i += 256) { a += loss[i]; b += loss3[i]; }
    s1[t] = a; s2[t] = b;
    __syncthreads();
    for (int off = 128; off > 0; off >>= 1) {
        if (t < off) { s1[t] += s1[t + off]; s2[t] += s2[t + off]; }
        __syncthreads();
    }
    if (t == 0) {
        float inv = 1.0f / (float)NROWS;
        out[0] = (s1[0] + s2[0]) * inv;
        out[1] = s1[0] * inv;
        out[2] = s2[0] * inv;
    }
}

// ---------------------------------------------------------------------------
extern "C" void kernel_launch(void* const* d_in, const int* in_sizes, int n_in,
                              void* d_out, int out_size, void* d_ws, size_t ws_size,
                              hipStream_t stream) {
    const float* X   = (const float*)d_in[0];
    const int*   tgt = (const int*)d_in[1];
    const int*   ind = (const int*)d_in[2];
    float* out = (float*)d_out;

    float* sq    = (float*)d_ws;
    float* dap   = sq + NROWS;
    float* dan   = dap + NROWS;
    float* loss  = dan + NROWS;
    float* loss3 = loss + NROWS;
    int*   aidx  = (int*)(loss3 + NROWS);

    sumsq_kernel   <<<NROWS / 256, 256, 0, stream>>>(X, sq);
    hardmine_kernel<<<NROWS / ROWS_PER_BLOCK, 256, 0, stream>>>(X, tgt, sq, dap, dan, aidx);
    pairloss_kernel<<<NROWS / 256, 256, 0, stream>>>(X, sq, ind, dap, dan, aidx, loss, loss3);
    finalize_kernel<<<1, 256, 0, stream>>>(loss, loss3, out);
}